// Net_76776835384059
// MI455X (gfx1250) — compile-verified
//
#include <hip/hip_runtime.h>
#include <hip/hip_bf16.h>

#define NN   262144
#define EE   1048576
#define GG   8192
#define F_IN 75
#define DD   64
#define BN_EPS 1e-5f

typedef __attribute__((ext_vector_type(2))) float v2f;
typedef __attribute__((ext_vector_type(8))) float v8f;

static inline unsigned cdiv(unsigned a, unsigned b) { return (a + b - 1u) / b; }

// ---------------------------------------------------------------------------
__global__ __launch_bounds__(256) void k_fill_f(float* p, float v, int n) {
    int i = blockIdx.x * 256 + threadIdx.x;
    if (i < n) p[i] = v;
}
__global__ __launch_bounds__(256) void k_fill_i(int* p, int v, int n) {
    int i = blockIdx.x * 256 + threadIdx.x;
    if (i < n) p[i] = v;
}
__global__ void k_set1_i(int* p, int v) { *p = v; }

__global__ __launch_bounds__(256) void k_count(const int* __restrict__ dst, int* cnt, int e) {
    int i = blockIdx.x * 256 + threadIdx.x;
    if (i < e) atomicAdd(&cnt[dst[i]], 1);
}
__global__ __launch_bounds__(256) void k_dinv(const int* __restrict__ cnt, float* dinv, int n) {
    int i = blockIdx.x * 256 + threadIdx.x;
    if (i < n) dinv[i] = rsqrtf(1.0f + (float)cnt[i]);
}

// ---------------------------------------------------------------------------
// Exclusive scan of N=262144 ints: 256 blocks x 1024 elems, then block offsets.
__global__ __launch_bounds__(256) void k_scan1(const int* __restrict__ in,
                                               int* __restrict__ out, int* __restrict__ bsum) {
    __shared__ int s[256];
    int tid = threadIdx.x;
    int4 v = ((const int4*)in)[blockIdx.x * 256 + tid];
    int t = v.x + v.y + v.z + v.w;
    s[tid] = t;
    __syncthreads();
    for (int off = 1; off < 256; off <<= 1) {
        int o = (tid >= off) ? s[tid - off] : 0;
        __syncthreads();
        s[tid] += o;
        __syncthreads();
    }
    int excl = s[tid] - t;
    int4 o; o.x = excl; o.y = excl + v.x; o.z = o.y + v.y; o.w = o.z + v.z;
    ((int4*)out)[blockIdx.x * 256 + tid] = o;
    if (tid == 255) bsum[blockIdx.x] = s[255];
}
__global__ void k_scan2(int* b) {
    __shared__ int s[256];
    int tid = threadIdx.x;
    int t = b[tid];
    s[tid] = t;
    __syncthreads();
    for (int off = 1; off < 256; off <<= 1) {
        int o = (tid >= off) ? s[tid - off] : 0;
        __syncthreads();
        s[tid] += o;
        __syncthreads();
    }
    b[tid] = s[tid] - t;
}
__global__ __launch_bounds__(256) void k_scan3(int* out, const int* __restrict__ boff, int n) {
    int i = blockIdx.x * 256 + threadIdx.x;
    if (i < n) out[i] += boff[i >> 10];
}
__global__ __launch_bounds__(256) void k_copy_i(const int* __restrict__ a, int* b, int n) {
    int i = blockIdx.x * 256 + threadIdx.x;
    if (i < n) b[i] = a[i];
}
__global__ __launch_bounds__(256) void k_csr_fill(const int* __restrict__ src,
                                                  const int* __restrict__ dst,
                                                  int* cursor, int* esrc, int e) {
    int i = blockIdx.x * 256 + threadIdx.x;
    if (i >= e) return;
    int p = atomicAdd(&cursor[dst[i]], 1);
    esrc[p] = src[i];
}

// ---------------------------------------------------------------------------
// Specialized WMMA fp32 GEMM for K=64: C[M,64] = A[M,64] @ W[64,64] + bias.
// B fragments for the whole W live in registers (64 x v2f = 128 VGPRs/wave),
// loaded from LDS once; grid-stride over 128-row M tiles amortizes the preload.
// Inner loop = 1 aligned b64 A-load + 4 v_wmma per K-step. M % 128 == 0.
__global__ __launch_bounds__(256) void k_gemm_k64(
    const float* __restrict__ A, const float* __restrict__ W,
    const float* __restrict__ bias, float* __restrict__ C,
    int M, int relu)
{
    __shared__ float Wl[64 * 64];
    const int tid = threadIdx.x;
    for (int idx = tid; idx < 64 * 64; idx += 256) Wl[idx] = W[idx];
    __syncthreads();

    const int wave = tid >> 5;
    const int lane = tid & 31;
    const int lrow = lane & 15;
    const int koff = (lane >> 4) << 1;     // lanes 16-31 hold K+2,K+3

    // Preload all B fragments into registers (once).
    v2f Bf[16][4];
    #pragma unroll
    for (int ks = 0; ks < 16; ++ks) {
        const float* w0 = &Wl[(ks * 4 + koff) * 64];
        const float* w1 = w0 + 64;
        #pragma unroll
        for (int nt = 0; nt < 4; ++nt) {
            Bf[ks][nt].x = w0[nt * 16 + lrow];
            Bf[ks][nt].y = w1[nt * 16 + lrow];
        }
    }

    const int mstride = gridDim.x * 128;
    for (int row0 = blockIdx.x * 128; row0 < M; row0 += mstride) {
        const float* Arow = A + (size_t)(row0 + wave * 16 + lrow) * 64 + koff;
        // prefetch next tile's A strip (global_prefetch_b8)
        if (row0 + mstride < M)
            __builtin_prefetch(Arow + (size_t)mstride * 64, 0, 1);

        v8f acc0 = {}, acc1 = {}, acc2 = {}, acc3 = {};
        #pragma unroll
        for (int ks = 0; ks < 16; ++ks) {
            v2f a = *(const v2f*)(Arow + ks * 4);   // 8B-aligned
            acc0 = __builtin_amdgcn_wmma_f32_16x16x4_f32(false, a, false, Bf[ks][0], (short)0, acc0, false, false);
            acc1 = __builtin_amdgcn_wmma_f32_16x16x4_f32(false, a, false, Bf[ks][1], (short)0, acc1, false, false);
            acc2 = __builtin_amdgcn_wmma_f32_16x16x4_f32(false, a, false, Bf[ks][2], (short)0, acc2, false, false);
            acc3 = __builtin_amdgcn_wmma_f32_16x16x4_f32(false, a, false, Bf[ks][3], (short)0, acc3, false, false);
        }

        const int rbase = row0 + wave * 16 + ((lane >> 4) * 8);
        #pragma unroll
        for (int nt = 0; nt < 4; ++nt) {
            v8f acc = (nt == 0) ? acc0 : (nt == 1) ? acc1 : (nt == 2) ? acc2 : acc3;
            const int col = nt * 16 + lrow;
            const float bv = bias ? bias[col] : 0.f;
            #pragma unroll
            for (int r = 0; r < 8; ++r) {
                float v = acc[r] + bv;
                if (relu) v = fmaxf(v, 0.f);
                C[(size_t)(rbase + r) * 64 + col] = v;
            }
        }
    }
}

// ---------------------------------------------------------------------------
// Generic WMMA fp32 GEMM (K=75 padded / K=128): C[M,64] = A[M,K] @ W[K,64] + bias
template <bool EXACT>
__global__ __launch_bounds__(256) void k_gemm_n64(
    const float* __restrict__ A, const float* __restrict__ W,
    const float* __restrict__ bias, float* __restrict__ C,
    int M, int K, int relu)
{
    __shared__ float Wl[128 * 64];
    const int tid = threadIdx.x;
    const int Kp  = (K + 3) & ~3;

    for (int idx = tid; idx < Kp * 64; idx += 256) {
        int r = idx >> 6, c = idx & 63;
        Wl[idx] = (r < K) ? W[r * 64 + c] : 0.f;
    }
    __syncthreads();

    const int wave = tid >> 5;
    const int lane = tid & 31;
    const int lrow = lane & 15;
    const int koff = (lane >> 4) << 1;
    const int row  = blockIdx.x * 128 + wave * 16 + lrow;
    const float* Arow = A + (size_t)row * K;

    v8f acc0 = {}, acc1 = {}, acc2 = {}, acc3 = {};

    for (int kb = 0; kb < Kp; kb += 4) {
        const int k0 = kb + koff, k1 = k0 + 1;
        float a0, a1;
        if (EXACT) {
            a0 = Arow[k0];
            a1 = Arow[k1];
        } else {
            int kc0 = k0 < K ? k0 : K - 1;   // loads always in-bounds,
            int kc1 = k1 < K ? k1 : K - 1;   // value zeroed by cndmask
            a0 = Arow[kc0];
            a1 = Arow[kc1];
            a0 = k0 < K ? a0 : 0.f;
            a1 = k1 < K ? a1 : 0.f;
        }
        v2f a; a.x = a0; a.y = a1;
        const float* w0 = &Wl[k0 * 64];
        const float* w1 = &Wl[k1 * 64];
        v2f b;
        b.x = w0[lrow];      b.y = w1[lrow];
        acc0 = __builtin_amdgcn_wmma_f32_16x16x4_f32(false, a, false, b, (short)0, acc0, false, false);
        b.x = w0[16 + lrow]; b.y = w1[16 + lrow];
        acc1 = __builtin_amdgcn_wmma_f32_16x16x4_f32(false, a, false, b, (short)0, acc1, false, false);
        b.x = w0[32 + lrow]; b.y = w1[32 + lrow];
        acc2 = __builtin_amdgcn_wmma_f32_16x16x4_f32(false, a, false, b, (short)0, acc2, false, false);
        b.x = w0[48 + lrow]; b.y = w1[48 + lrow];
        acc3 = __builtin_amdgcn_wmma_f32_16x16x4_f32(false, a, false, b, (short)0, acc3, false, false);
    }

    const int rhalf = (lane >> 4) * 8;
    const int rbase = blockIdx.x * 128 + wave * 16 + rhalf;
    #pragma unroll
    for (int nt = 0; nt < 4; ++nt) {
        v8f acc = (nt == 0) ? acc0 : (nt == 1) ? acc1 : (nt == 2) ? acc2 : acc3;
        const int col = nt * 16 + lrow;
        const float bv = bias ? bias[col] : 0.f;
        #pragma unroll
        for (int r = 0; r < 8; ++r) {
            float v = acc[r] + bv;
            if (relu) v = fmaxf(v, 0.f);
            C[(size_t)(rbase + r) * 64 + col] = v;
        }
    }
}

// ---------------------------------------------------------------------------
// Fold next-layer BN affine into W and a per-column bias:
//   (a*sc_k + sh_k) @ W = a @ (sc_k*W[k,:]) + sum_k sh_k*W[k,:]
__global__ void k_fold(const float* __restrict__ W, const float* __restrict__ sc,
                       const float* __restrict__ sh, const float* __restrict__ base,
                       float* __restrict__ Wsc, float* __restrict__ foldb)
{
    int n = threadIdx.x;  // 64 threads
    float acc = base ? base[n] : 0.f;
    for (int k = 0; k < 64; ++k) {
        float w = W[k * 64 + n];
        Wsc[k * 64 + n] = w * sc[k];
        acc = fmaf(sh[k], w, acc);
    }
    foldb[n] = acc;
}

// ---------------------------------------------------------------------------
// Atomic-free GCN aggregation via CSR (in-edges grouped by dst):
// AGG[n,:] = dinv[n]*sum_{s in nbr(n)} H[s,:]*dinv[s] + H[n,:]*dinv[n]^2 + b
__global__ __launch_bounds__(256) void k_gcn_agg(
    const float* __restrict__ H, const float* __restrict__ dinv,
    const int* __restrict__ rowptr, const int* __restrict__ esrc,
    const float* __restrict__ bias, float* __restrict__ AGG, int total)
{
    int gid = blockIdx.x * 256 + threadIdx.x;
    if (gid >= total) return;
    int n  = gid >> 4;
    int c4 = (gid & 15) << 2;
    float din = dinv[n];
    int beg = rowptr[n], end = rowptr[n + 1];

    float4 acc = make_float4(0.f, 0.f, 0.f, 0.f);
    for (int j = beg; j < end; ++j) {
        int s = esrc[j];
        float w = dinv[s];
        float4 h = *(const float4*)(H + (size_t)s * 64 + c4);
        acc.x = fmaf(h.x, w, acc.x);
        acc.y = fmaf(h.y, w, acc.y);
        acc.z = fmaf(h.z, w, acc.z);
        acc.w = fmaf(h.w, w, acc.w);
    }
    float4 hs = *(const float4*)(H + (size_t)n * 64 + c4);
    float d2 = din * din;
    float4 o;
    o.x = fmaf(acc.x, din, fmaf(hs.x, d2, bias[c4 + 0]));
    o.y = fmaf(acc.y, din, fmaf(hs.y, d2, bias[c4 + 1]));
    o.z = fmaf(acc.z, din, fmaf(hs.z, d2, bias[c4 + 2]));
    o.w = fmaf(acc.w, din, fmaf(hs.w, d2, bias[c4 + 3]));
    *(float4*)(AGG + (size_t)n * 64 + c4) = o;
}

// ---------------------------------------------------------------------------
__global__ __launch_bounds__(256) void k_relu_stats(
    float* __restrict__ Y, float* __restrict__ stats, int total4)
{
    __shared__ float ls[128];
    int tid = threadIdx.x;
    if (tid < 128) ls[tid] = 0.f;
    __syncthreads();
    int gid = blockIdx.x * 256 + tid;
    if (gid < total4) {
        int c4 = (gid & 15) << 2;
        float4 v = ((float4*)Y)[gid];
        v.x = fmaxf(v.x, 0.f); v.y = fmaxf(v.y, 0.f);
        v.z = fmaxf(v.z, 0.f); v.w = fmaxf(v.w, 0.f);
        ((float4*)Y)[gid] = v;
        atomicAdd(&ls[c4 + 0], v.x);       atomicAdd(&ls[64 + c4 + 0], v.x * v.x);
        atomicAdd(&ls[c4 + 1], v.y);       atomicAdd(&ls[64 + c4 + 1], v.y * v.y);
        atomicAdd(&ls[c4 + 2], v.z);       atomicAdd(&ls[64 + c4 + 2], v.z * v.z);
        atomicAdd(&ls[c4 + 3], v.w);       atomicAdd(&ls[64 + c4 + 3], v.w * v.w);
    }
    __syncthreads();
    if (tid < 128) atomicAdd(&stats[tid], ls[tid]);
}

__global__ void k_bn_finalize(
    const float* __restrict__ stats, const float* __restrict__ gamma,
    const float* __restrict__ beta, float* __restrict__ scale,
    float* __restrict__ shift, float invN)
{
    int c = threadIdx.x;  // 64
    float m   = stats[c] * invN;
    float var = stats[64 + c] * invN - m * m;
    float s   = gamma[c] * rsqrtf(var + BN_EPS);
    scale[c] = s;
    shift[c] = beta[c] - m * s;
}

__global__ __launch_bounds__(256) void k_pool(
    const float* __restrict__ Y, const float* __restrict__ scale,
    const float* __restrict__ shift, const int* __restrict__ batch,
    float* __restrict__ Z, int total4)
{
    int gid = blockIdx.x * 256 + threadIdx.x;
    if (gid >= total4) return;
    int n  = gid >> 4;
    int c4 = (gid & 15) << 2;
    float4 v = ((const float4*)Y)[gid];
    float4 sc = *(const float4*)(scale + c4);
    float4 sh = *(const float4*)(shift + c4);
    float* z = Z + (size_t)batch[n] * 64 + c4;
    atomicAdd(z + 0, fmaf(v.x, sc.x, sh.x));
    atomicAdd(z + 1, fmaf(v.y, sc.y, sh.y));
    atomicAdd(z + 2, fmaf(v.z, sc.z, sh.z));
    atomicAdd(z + 3, fmaf(v.w, sc.w, sh.w));
}

__global__ __launch_bounds__(256) void k_concat(
    const float* __restrict__ Z1, const float* __restrict__ Z2,
    float* __restrict__ Zc, int total)
{
    int gid = blockIdx.x * 256 + threadIdx.x;
    if (gid >= total) return;
    int g = gid >> 7, c = gid & 127;
    Zc[gid] = (c < 64) ? Z1[g * 64 + c] : Z2[g * 64 + (c - 64)];
}

__global__ __launch_bounds__(256) void k_fc_out(
    const float* __restrict__ H3, const float* __restrict__ W4,
    const float* __restrict__ b4, float* __restrict__ out, int g)
{
    int i = blockIdx.x * 256 + threadIdx.x;
    if (i >= g) return;
    float acc = b4[0];
    const float* h = H3 + (size_t)i * 64;
    #pragma unroll
    for (int k = 0; k < 64; ++k) acc = fmaf(h[k], W4[k], acc);
    out[i] = acc;
}

// ---------------------------------------------------------------------------
extern "C" void kernel_launch(void* const* d_in, const int* in_sizes, int n_in,
                              void* d_out, int out_size, void* d_ws, size_t ws_size,
                              hipStream_t stream)
{
    (void)in_sizes; (void)n_in; (void)out_size; (void)ws_size;

    // Inputs flattened in setup_inputs() insertion order (50 leaves)
    const float* x1 = (const float*)d_in[0];
    const int*   e1 = (const int*)  d_in[1];
    const int*   b1 = (const int*)  d_in[2];
    const float* x2 = (const float*)d_in[3];
    const int*   e2 = (const int*)  d_in[4];
    const int*   b2 = (const int*)  d_in[5];
    const float* p1[16]; for (int i = 0; i < 16; ++i) p1[i] = (const float*)d_in[6 + i];
    const float* p2[16]; for (int i = 0; i < 16; ++i) p2[i] = (const float*)d_in[22 + i];
    const float* pf[12]; for (int i = 0; i < 12; ++i) pf[i] = (const float*)d_in[38 + i];

    // Workspace carving (16B-aligned sections)
    float* ws    = (float*)d_ws;
    float* H     = ws;                               // N*64
    float* Y     = H     + (size_t)NN * 64;          // N*64
    float* dinv  = Y     + (size_t)NN * 64;          // N
    float* Z1    = dinv  + NN;                       // G*64
    float* Z2    = Z1    + (size_t)GG * 64;          // G*64
    float* Zc    = Z2    + (size_t)GG * 64;          // G*128
    float* F1    = Zc    + (size_t)GG * 128;         // G*64
    float* F2    = F1    + (size_t)GG * 64;          // G*64
    float* stats = F2    + (size_t)GG * 64;          // 128
    float* scale = stats + 128;                      // 64
    float* shift = scale + 64;                       // 64
    float* Wsc   = shift + 64;                       // 64*64
    float* foldb = Wsc   + 64 * 64;                  // 64
    int*   cnt    = (int*)(foldb + 64);              // N
    int*   rowptr = cnt    + NN;                     // N+1 (pad to N+4)
    int*   cursor = rowptr + NN + 4;                 // N
    int*   esrc   = cursor + NN;                     // E
    int*   bsum   = esrc   + EE;                     // 256

    const float invN = 1.0f / (float)NN;
    const float invG = 1.0f / (float)GG;

    auto run_branch = [&](const float* X0, const int* edge, const int* batch,
                          const float* const* P, float* Z) {
        const int* src = edge;
        const int* dst = edge + EE;

        // --- build CSR (in-edges grouped by dst) + dinv, once per branch ---
        k_fill_i<<<cdiv(NN, 256), 256, 0, stream>>>(cnt, 0, NN);
        k_count<<<cdiv(EE, 256), 256, 0, stream>>>(dst, cnt, EE);
        k_dinv<<<cdiv(NN, 256), 256, 0, stream>>>(cnt, dinv, NN);
        k_scan1<<<NN / 1024, 256, 0, stream>>>(cnt, rowptr, bsum);
        k_scan2<<<1, 256, 0, stream>>>(bsum);
        k_scan3<<<cdiv(NN, 256), 256, 0, stream>>>(rowptr, bsum, NN);
        k_set1_i<<<1, 1, 0, stream>>>(rowptr + NN, EE);
        k_copy_i<<<cdiv(NN, 256), 256, 0, stream>>>(rowptr, cursor, NN);
        k_csr_fill<<<cdiv(EE, 256), 256, 0, stream>>>(src, dst, cursor, esrc, EE);

        // --- 4 GCN layers ---
        const float* curA = X0;
        const float* curW = P[0];
        const float* gbias = nullptr;     // per-column bias folded from BN shift
        for (int l = 0; l < 4; ++l) {
            const float* bias  = P[l * 4 + 1];
            const float* gamma = P[l * 4 + 2];
            const float* beta  = P[l * 4 + 3];
            if (l == 0)
                k_gemm_n64<false><<<NN / 128, 256, 0, stream>>>(curA, curW, gbias, H, NN, F_IN, 0);
            else
                k_gemm_k64<<<1024, 256, 0, stream>>>(curA, curW, gbias, H, NN, 0);
            // aggregation (self-loop + bias fused), no atomics
            k_gcn_agg<<<cdiv(NN * 16, 256), 256, 0, stream>>>(H, dinv, rowptr, esrc, bias, Y, NN * 16);
            // ReLU + BN batch statistics
            k_fill_f<<<1, 128, 0, stream>>>(stats, 0.f, 128);
            k_relu_stats<<<cdiv(NN * 16, 256), 256, 0, stream>>>(Y, stats, NN * 16);
            k_bn_finalize<<<1, 64, 0, stream>>>(stats, gamma, beta, scale, shift, invN);
            if (l < 3) {  // fold BN affine into next layer's W + column bias
                k_fold<<<1, 64, 0, stream>>>(P[(l + 1) * 4], scale, shift, nullptr, Wsc, foldb);
                curA = Y; curW = Wsc; gbias = foldb;
            }
        }
        // global_add_pool with fused final-layer BN affine
        k_fill_f<<<cdiv(GG * 64, 256), 256, 0, stream>>>(Z, 0.f, GG * 64);
        k_pool<<<cdiv(NN * 16, 256), 256, 0, stream>>>(Y, scale, shift, batch, Z, NN * 16);
    };

    run_branch(x1, e1, b1, p1, Z1);
    run_branch(x2, e2, b2, p2, Z2);

    // --- FC head ---
    k_concat<<<cdiv(GG * 128, 256), 256, 0, stream>>>(Zc ? Z1 : Z1, Z2, Zc, GG * 128);
    // fc1: relu(Zc @ fW1 + fb1) -> BN stats
    k_gemm_n64<true><<<GG / 128, 256, 0, stream>>>(Zc, pf[0], pf[1], F1, GG, 128, 0);
    k_fill_f<<<1, 128, 0, stream>>>(stats, 0.f, 128);
    k_relu_stats<<<cdiv(GG * 16, 256), 256, 0, stream>>>(F1, stats, GG * 16);
    k_bn_finalize<<<1, 64, 0, stream>>>(stats, pf[2], pf[3], scale, shift, invG);
    // fc2: relu(BN(F1) @ fW2 + fb2) -> BN stats   (BN folded into W)
    k_fold<<<1, 64, 0, stream>>>(pf[4], scale, shift, pf[5], Wsc, foldb);
    k_gemm_k64<<<GG / 128, 256, 0, stream>>>(F1, Wsc, foldb, F2, GG, 0);
    k_fill_f<<<1, 128, 0, stream>>>(stats, 0.f, 128);
    k_relu_stats<<<cdiv(GG * 16, 256), 256, 0, stream>>>(F2, stats, GG * 16);
    k_bn_finalize<<<1, 64, 0, stream>>>(stats, pf[6], pf[7], scale, shift, invG);
    // fc3: relu(BN(F2) @ fW3 + fb3)   (ReLU fused in epilogue)
    k_fold<<<1, 64, 0, stream>>>(pf[8], scale, shift, pf[9], Wsc, foldb);
    k_gemm_k64<<<GG / 128, 256, 0, stream>>>(F2, Wsc, foldb, F1, GG, 1);
    // fc4
    k_fc_out<<<cdiv(GG, 256), 256, 0, stream>>>(F1, pf[10], pf[11], (float*)d_out, GG);
}